// SAGE_3547642987366
// MI455X (gfx1250) — compile-verified
//
#include <hip/hip_runtime.h>
#include <hip/hip_bf16.h>

typedef __attribute__((ext_vector_type(2))) float v2f;
typedef __attribute__((ext_vector_type(8))) float v8f;

#define N0c 260000
#define N1c 60000
#define N2c 10000
#define E1c 1500000
#define E2c 250000
#define DINc 128
#define DHc 256
#define DOUTc 128

// ---------------------------------------------------------------------------
// Degree counts for both layers in one launch (f32 atomics into L2).
// ---------------------------------------------------------------------------
__global__ void sage_count_kernel(const int* __restrict__ dst1, float* __restrict__ cnt1,
                                  const int* __restrict__ dst2, float* __restrict__ cnt2) {
    int i = blockIdx.x * blockDim.x + threadIdx.x;
    if (i < E1c) {
        atomicAdd(&cnt1[dst1[i]], 1.0f);
    } else if (i < E1c + E2c) {
        atomicAdd(&cnt2[dst2[i - E1c]], 1.0f);
    }
}

// ---------------------------------------------------------------------------
// Scatter-add of source rows into per-target accumulators.
// One thread = one float4 chunk of one edge's feature row. For D=128 a wave
// (32 lanes) covers exactly one 512B row -> fully coalesced gather; the
// feature table is L2-resident (133MB < 192MB L2).
// ---------------------------------------------------------------------------
__global__ void sage_scatter_kernel(const float* __restrict__ xsrc,
                                    const int* __restrict__ src,
                                    const int* __restrict__ dst,
                                    float* __restrict__ agg,
                                    int E, int D4) {
    long long idx = (long long)blockIdx.x * blockDim.x + threadIdx.x;
    if (idx >= (long long)E * D4) return;
    int e = (int)(idx / D4);
    int c = (int)(idx - (long long)e * D4);
    int s = src[e];
    int d = dst[e];
    const float4 v = reinterpret_cast<const float4*>(xsrc + (size_t)s * D4 * 4)[c];
    float* ap = agg + (size_t)d * D4 * 4 + (size_t)c * 4;
    atomicAdd(ap + 0, v.x);
    atomicAdd(ap + 1, v.y);
    atomicAdd(ap + 2, v.z);
    atomicAdd(ap + 3, v.w);
}

// ---------------------------------------------------------------------------
// Fused SAGE linear stage via V_WMMA_F32_16X16X4_F32 (full f32 precision):
//   out = [relu]( (Aagg * 1/max(cnt,1)) @ Wl + bias + Aself @ Wr )
// One wave computes one 16x16 output tile, looping K in steps of 4.
// A-fragment layout (ISA 7.12.2, 32-bit A 16x4): lanes 0-15 hold K=k0..k0+1
// in v[0..1], lanes 16-31 hold K=k0+2..k0+3. B striped over N across lanes.
// Tile grids divide exactly so EXEC is all-1s at every WMMA.
// ---------------------------------------------------------------------------
__global__ __launch_bounds__(256) void sage_gemm_wmma_kernel(
    const float* __restrict__ Aagg,   // [M,K] un-normalized aggregate
    const float* __restrict__ cnt,    // [M] degree counts
    const float* __restrict__ Wl,     // [K,N]
    const float* __restrict__ Aself,  // [M,K] target-node features
    const float* __restrict__ Wr,     // [K,N]
    const float* __restrict__ bias,   // [N]
    float* __restrict__ out,          // [M,N]
    int M, int N, int K, int ntiles_n, int do_relu) {
    int wave = threadIdx.x >> 5;
    int lane = threadIdx.x & 31;
    int tile = blockIdx.x * 8 + wave;
    int mt = tile / ntiles_n;
    int nt = tile - mt * ntiles_n;
    if (mt * 16 >= M) return;  // wave-uniform; never taken with exact grids

    const int m0 = mt * 16;
    const int n0 = nt * 16;
    const int half = lane >> 4;   // 0: K=k0..k0+1, 1: K=k0+2..k0+3
    const int l = lane & 15;
    const int rowA = m0 + l;

    const float inv = 1.0f / fmaxf(cnt[rowA], 1.0f);
    const float* aBase = Aagg + (size_t)rowA * K + 2 * half;
    const float* sBase = Aself + (size_t)rowA * K + 2 * half;

    v8f c = {};

    // agg @ Wl  (degree normalization folded into the A fragment)
    for (int k0 = 0; k0 < K; k0 += 4) {
        v2f a;
        a.x = aBase[k0] * inv;
        a.y = aBase[k0 + 1] * inv;
        const int kb = k0 + 2 * half;
        v2f b;
        b.x = Wl[(size_t)kb * N + n0 + l];
        b.y = Wl[(size_t)(kb + 1) * N + n0 + l];
        c = __builtin_amdgcn_wmma_f32_16x16x4_f32(false, a, false, b,
                                                  (short)0, c, false, false);
    }
    // x_tgt @ Wr
    for (int k0 = 0; k0 < K; k0 += 4) {
        v2f a;
        a.x = sBase[k0];
        a.y = sBase[k0 + 1];
        const int kb = k0 + 2 * half;
        v2f b;
        b.x = Wr[(size_t)kb * N + n0 + l];
        b.y = Wr[(size_t)(kb + 1) * N + n0 + l];
        c = __builtin_amdgcn_wmma_f32_16x16x4_f32(false, a, false, b,
                                                  (short)0, c, false, false);
    }

    const float bv = bias[n0 + l];
    // C layout: VGPR r -> row m0 + r + 8*half, col n0 + l
#pragma unroll
    for (int r = 0; r < 8; ++r) {
        float v = c[r] + bv;
        if (do_relu) v = fmaxf(v, 0.0f);
        out[(size_t)(m0 + r + 8 * half) * N + (n0 + l)] = v;
    }
}

extern "C" void kernel_launch(void* const* d_in, const int* in_sizes, int n_in,
                              void* d_out, int out_size, void* d_ws, size_t ws_size,
                              hipStream_t stream) {
    const float* x   = (const float*)d_in[0];   // [N0, DIN]
    const float* W1l = (const float*)d_in[1];   // [DIN, DH]
    const float* b1  = (const float*)d_in[2];   // [DH]
    const float* W1r = (const float*)d_in[3];   // [DIN, DH]
    const float* W2l = (const float*)d_in[4];   // [DH, DOUT]
    const float* b2  = (const float*)d_in[5];   // [DOUT]
    const float* W2r = (const float*)d_in[6];   // [DH, DOUT]
    const int* src1  = (const int*)d_in[7];     // [E1]
    const int* dst1  = (const int*)d_in[8];     // [E1]
    const int* src2  = (const int*)d_in[9];     // [E2]
    const int* dst2  = (const int*)d_in[10];    // [E2]
    float* out = (float*)d_out;                 // [N2, DOUT]

    // Workspace layout (floats)
    float* agg1 = (float*)d_ws;                         // N1*DIN
    float* cnt1 = agg1 + (size_t)N1c * DINc;            // N1
    float* h    = cnt1 + N1c;                           // N1*DH
    float* agg2 = h + (size_t)N1c * DHc;                // N2*DH
    float* cnt2 = agg2 + (size_t)N2c * DHc;             // N2

    // Zero the accumulators (agg1+cnt1 contiguous, agg2+cnt2 contiguous)
    hipMemsetAsync(agg1, 0, ((size_t)N1c * DINc + N1c) * sizeof(float), stream);
    hipMemsetAsync(agg2, 0, ((size_t)N2c * DHc + N2c) * sizeof(float), stream);

    // Degree counts for both layers
    {
        int total = E1c + E2c;
        int blocks = (total + 255) / 256;
        sage_count_kernel<<<blocks, 256, 0, stream>>>(dst1, cnt1, dst2, cnt2);
    }

    // Layer 1 scatter: E1 * (DIN/4) threads = 48M, exact grid
    sage_scatter_kernel<<<(E1c * (DINc / 4)) / 256, 256, 0, stream>>>(
        x, src1, dst1, agg1, E1c, DINc / 4);

    // Layer 1 fused GEMM + bias + relu -> h [N1, DH]
    // tiles: (60000/16) * (256/16) = 3750*16 = 60000; 8 waves/block -> 7500 blocks
    sage_gemm_wmma_kernel<<<(N1c / 16) * (DHc / 16) / 8, 256, 0, stream>>>(
        agg1, cnt1, W1l, x, W1r, b1, h, N1c, DHc, DINc, DHc / 16, 1);

    // Layer 2 scatter: E2 * (DH/4) threads = 16M, exact grid
    sage_scatter_kernel<<<(E2c * (DHc / 4)) / 256, 256, 0, stream>>>(
        h, src2, dst2, agg2, E2c, DHc / 4);

    // Layer 2 fused GEMM + bias -> out [N2, DOUT]
    // tiles: (10000/16) * (128/16) = 625*8 = 5000; 8 waves/block -> 625 blocks
    sage_gemm_wmma_kernel<<<(N2c / 16) * (DOUTc / 16) / 8, 256, 0, stream>>>(
        agg2, cnt2, W2l, h, W2r, b2, out, N2c, DOUTc, DHc, DOUTc / 16, 0);
}